// Amortized_VI_41360535060909
// MI455X (gfx1250) — compile-verified
//
#include <hip/hip_runtime.h>
#include <hip/hip_bf16.h>

// CDNA5 / gfx1250, wave32. WMMA f32 16x16x4 for the batched tiny-K MLP GEMMs.
// Weights pre-staged into LDS in WMMA-B layout (zero padding baked in) so the
// hot loop has zero divergent loads: B operands are single ds_load_b64 each.

typedef float v2f __attribute__((ext_vector_type(2)));
typedef float v8f __attribute__((ext_vector_type(8)));

#define LDS_STRIDE 36     // 16x36 f32 staging per wave (pad vs 64 banks)
#define TILES_PER_WAVE 4  // persistent: each wave handles 4 x 16 rows

struct Head { const float *W1, *b1, *W2, *b2, *W3, *b3; };

static __device__ __forceinline__ v8f wmma_k4(v2f a, v2f b, v8f c) {
  // D = A(16x4,f32) x B(4x16,f32) + C(16x16,f32)
  return __builtin_amdgcn_wmma_f32_16x16x4_f32(
      /*neg_a=*/false, a, /*neg_b=*/false, b,
      /*c_mod=*/(short)0, c, /*reuse_a=*/false, /*reuse_b=*/false);
}

static __device__ __forceinline__ void relu8(v8f &c) {
#pragma unroll
  for (int i = 0; i < 8; ++i) c[i] = fmaxf(c[i], 0.0f);
}

static __device__ __forceinline__ float softplus_f(float x) {
  return fmaxf(x, 0.0f) + __logf(1.0f + __expf(-fabsf(x)));
}

static __device__ __forceinline__ Head head_sel(int h, Head a, Head b, Head c) {
  Head r;
  r.W1 = h == 0 ? a.W1 : (h == 1 ? b.W1 : c.W1);
  r.b1 = h == 0 ? a.b1 : (h == 1 ? b.b1 : c.b1);
  r.W2 = h == 0 ? a.W2 : (h == 1 ? b.W2 : c.W2);
  r.b2 = h == 0 ? a.b2 : (h == 1 ? b.b2 : c.b2);
  r.W3 = h == 0 ? a.W3 : (h == 1 ? b.W3 : c.W3);
  r.b3 = h == 0 ? a.b3 : (h == 1 ? b.b3 : c.b3);
  return r;
}

// One MLP head for a wave's 16-row tile.
// Bt:   10 pre-staged B tiles for this head, [tile][lane][2] in LDS.
// bias: 64 padded bias slots: [0..31]=b1, [32..47]=b2, [48..63]=b3.
// wbuf: this wave's private LDS staging region (16 x LDS_STRIDE floats).
static __device__ __forceinline__ v8f mlp_head(const float (*Bt)[32][2],
                                               const float* bias, float* wbuf,
                                               v2f a_y, int lane) {
  const int n    = lane & 15;   // tile column (B/C) == tile row M (A)
  const int half = lane >> 4;   // lanes 16-31: C rows 8..15, A/B K-pair {2,3}
  const int koff = half * 2;

  // ---- layer 1: [16x2] @ [2x20] (+b1), K padded 2->4, N padded 20->32
  v8f h1[2];
#pragma unroll
  for (int j = 0; j < 2; ++j) {
    const v2f b  = *(const v2f*)&Bt[j][lane][0];
    const float bb = bias[j * 16 + n];
    v8f c;
#pragma unroll
    for (int k = 0; k < 8; ++k) c[k] = bb;
    c = wmma_k4(a_y, b, c);
    relu8(c);
    h1[j] = c;
  }
  __syncthreads();
#pragma unroll
  for (int j = 0; j < 2; ++j)
#pragma unroll
    for (int k = 0; k < 8; ++k)
      wbuf[(k + 8 * half) * LDS_STRIDE + j * 16 + n] = h1[j][k];
  __syncthreads();

  // ---- layer 2: [16x20] @ [20x10] (+b2), K=20 as 5 chunks of 4
  {
    const float bb = bias[32 + n];
    v8f c;
#pragma unroll
    for (int k = 0; k < 8; ++k) c[k] = bb;
#pragma unroll
    for (int t = 0; t < 5; ++t) {
      const int kk = 4 * t + koff;                       // always even -> 8B aligned
      const v2f a = *(const v2f*)&wbuf[n * LDS_STRIDE + kk];
      const v2f b = *(const v2f*)&Bt[2 + t][lane][0];
      c = wmma_k4(a, b, c);
    }
    relu8(c);
    __syncthreads();
#pragma unroll
    for (int k = 0; k < 8; ++k)
      wbuf[(k + 8 * half) * LDS_STRIDE + n] = c[k];      // h2 cols 0..15 (10..15 == 0)
    __syncthreads();
  }

  // ---- layer 3: [16x10] @ [10xout] (+b3), K=10 padded to 12 (3 chunks)
  {
    const float bb = bias[48 + n];
    v8f c;
#pragma unroll
    for (int k = 0; k < 8; ++k) c[k] = bb;
#pragma unroll
    for (int t = 0; t < 3; ++t) {
      const int kk = 4 * t + koff;
      const v2f a = *(const v2f*)&wbuf[n * LDS_STRIDE + kk];  // cols 10,11 are 0
      const v2f b = *(const v2f*)&Bt[7 + t][lane][0];
      c = wmma_k4(a, b, c);
    }
    return c;
  }
}

__global__ __launch_bounds__(256) void avi_kernel(
    const float* __restrict__ y, const float* __restrict__ zs,
    Head mu, Head ld, Head lo, float* __restrict__ out, float invN) {
  __shared__ __align__(16) float sB[30][32][2];        // [head*10+tile][lane][2]
  __shared__ float sBias[3][64];                       // padded b1|b2|b3 per head
  __shared__ __align__(16) float wbuf_all[8][16 * LDS_STRIDE];
  __shared__ float obuf[8][16][48];                    // head outputs, unconditional
  __shared__ float red[256];

  const int tid  = threadIdx.x;
  const int wave = tid >> 5;
  const int lane = tid & 31;
  const int n    = lane & 15;
  const int half = lane >> 4;
  const int koff = half * 2;
  float* wbuf = wbuf_all[wave];

  // ================= weight preload (once per block, branch-free loads) ====
  // B tiles: pair p = head*10 + t; t: 0-1 layer1, 2-6 layer2, 7-9 layer3.
  for (int p = wave; p < 30; p += 8) {
    const int h = p / 10, t = p % 10;
    const Head H = head_sel(h, mu, ld, lo);
    const int od = (h == 2) ? 6 : 4;
    float v0, v1;
    if (t < 2) {                       // layer 1 tile j=t : rows {0,1} / pad {2,3}
      const int nn  = t * 16 + n;
      const bool vv = (half == 0) && (nn < 20);
      const int  ix = nn < 20 ? nn : 19;
      v0 = vv ? H.W1[ix] : 0.0f;       // W1[0][nn]
      v1 = vv ? H.W1[20 + ix] : 0.0f;  // W1[1][nn]
    } else if (t < 7) {                // layer 2 chunk q=t-2 : K rows 4q+koff(+1)
      const int kk  = 4 * (t - 2) + koff;      // <= 19, always valid
      const bool vv = (n < 10);
      const int  ix = n < 10 ? n : 9;
      v0 = vv ? H.W2[kk * 10 + ix] : 0.0f;
      v1 = vv ? H.W2[(kk + 1) * 10 + ix] : 0.0f;
    } else {                           // layer 3 chunk q=t-7 : K rows 4q+koff(+1)
      const int kk  = 4 * (t - 7) + koff;
      const bool nv = (n < od);
      const int  ix = nv ? n : (od - 1);
      const int  k0 = kk < 10 ? kk : 9;
      const int  k1 = (kk + 1) < 10 ? (kk + 1) : 9;
      v0 = (nv && kk < 10)     ? H.W3[k0 * od + ix] : 0.0f;
      v1 = (nv && (kk + 1) < 10) ? H.W3[k1 * od + ix] : 0.0f;
    }
    sB[p][lane][0] = v0;
    sB[p][lane][1] = v1;
  }
  // biases: 3 heads x 64 slots
  if (tid < 192) {
    const int h = tid >> 6, s = tid & 63;
    const Head H = head_sel(h, mu, ld, lo);
    const int od = (h == 2) ? 6 : 4;
    float v;
    if (s < 32)      v = (s < 20) ? H.b1[s < 20 ? s : 19] : 0.0f;
    else if (s < 48) { const int c0 = s - 32; v = (c0 < 10) ? H.b2[c0 < 10 ? c0 : 9] : 0.0f; }
    else             { const int c0 = s - 48; v = (c0 < od) ? H.b3[c0 < od ? c0 : od - 1] : 0.0f; }
    sBias[h][s] = v;
  }
  __syncthreads();

  const float4* z4 = reinterpret_cast<const float4*>(zs);
  float total = 0.0f;

  // ======================= persistent tile loop ============================
  for (int it = 0; it < TILES_PER_WAVE; ++it) {
    const int tile = (blockIdx.x * 8 + wave) * TILES_PER_WAVE + it;
    const int g    = tile * 16 + n;

    // warm the z stream for this tile (global_prefetch_b8)
    __builtin_prefetch(zs + (size_t)g * 32, 0, 1);

    // y loaded once: feeds the layer-1 A tile AND the likelihood residual.
    const float2 yv = *reinterpret_cast<const float2*>(y + 2 * (size_t)g);
    v2f a_y;                        // lanes 16-31 carry K=2,3 zero padding
    a_y[0] = half ? 0.0f : yv.x;
    a_y[1] = half ? 0.0f : yv.y;

    const v8f c_mu = mlp_head(&sB[0],  sBias[0], wbuf, a_y, lane);
#pragma unroll
    for (int k = 0; k < 8; ++k) obuf[wave][k + 8 * half][n]      = c_mu[k];
    const v8f c_ld = mlp_head(&sB[10], sBias[1], wbuf, a_y, lane);
#pragma unroll
    for (int k = 0; k < 8; ++k) obuf[wave][k + 8 * half][16 + n] = c_ld[k];
    const v8f c_lo = mlp_head(&sB[20], sBias[2], wbuf, a_y, lane);
#pragma unroll
    for (int k = 0; k < 8; ++k) obuf[wave][k + 8 * half][32 + n] = c_lo[k];
    __syncthreads();

    // ---- sampling / likelihood: 2 lanes per row, 4 samples each ----------
    float muv[4], qld[4], lov[6];
#pragma unroll
    for (int i = 0; i < 4; ++i) muv[i] = obuf[wave][n][i];
#pragma unroll
    for (int i = 0; i < 4; ++i) qld[i] = softplus_f(obuf[wave][n][16 + i]);
#pragma unroll
    for (int j = 0; j < 6; ++j) lov[j] = obuf[wave][n][32 + j];

    // entropy: 0.5*4*(1+log2pi) + sum log diag(L)
    float ent = 5.67575414f;
#pragma unroll
    for (int i = 0; i < 4; ++i) ent += __logf(qld[i]);

    float acc = 0.0f;
#pragma unroll
    for (int s = 0; s < 4; ++s) {
      const float4 z = z4[(size_t)g * 8 + half * 4 + s];
      // xi = mu + L @ z   (L lower-triangular; np.tril_indices(4,-1) order)
      const float xi0 = muv[0] + qld[0] * z.x;
      const float xi1 = muv[1] + lov[0] * z.x + qld[1] * z.y;
      const float xi2 = muv[2] + lov[1] * z.x + lov[2] * z.y + qld[2] * z.z;
      const float xi3 = muv[3] + lov[3] * z.x + lov[4] * z.y + lov[5] * z.z + qld[3] * z.w;
      // log prior: Normal(0, {0.25,0.5,0.5,0.5})
      const float t0 = xi0 * 4.0f, t1 = xi1 * 2.0f, t2 = xi2 * 2.0f, t3 = xi3 * 2.0f;
      const float lp = -0.5f * (t0 * t0 + t1 * t1 + t2 * t2 + t3 * t3) - 0.21001822f;
      // forward kinematics
      const float a1 = xi1, a2 = a1 + xi2, a3 = a2 + xi3;
      const float px = 0.5f * __cosf(a1) + 0.5f * __cosf(a2) + __cosf(a3);
      const float py = xi0 + 0.5f * __sinf(a1) + 0.5f * __sinf(a2) + __sinf(a3);
      // log lik: Normal(D, 0.01)
      const float r0 = (yv.x - px) * 100.0f;
      const float r1 = (yv.y - py) * 100.0f;
      acc += lp - 0.5f * (r0 * r0 + r1 * r1) + 7.37246330f;
    }
    // per-row: mean over P samples + entropy (counted once per row)
    total += acc * 0.125f + (half == 0 ? ent : 0.0f);
    __syncthreads();   // obuf/wbuf safe to reuse next iteration
  }

  // block reduction -> single f32 atomic per block
  red[tid] = total * invN;
  __syncthreads();
#pragma unroll
  for (int off = 128; off > 0; off >>= 1) {
    if (tid < off) red[tid] += red[tid + off];
    __syncthreads();
  }
  if (tid == 0) atomicAdd(out, red[0]);
}

extern "C" void kernel_launch(void* const* d_in, const int* in_sizes, int n_in,
                              void* d_out, int out_size, void* d_ws, size_t ws_size,
                              hipStream_t stream) {
  (void)n_in; (void)d_ws; (void)ws_size;
  const float* y  = (const float*)d_in[0];
  const float* zs = (const float*)d_in[1];
  Head mu{(const float*)d_in[2],  (const float*)d_in[3],  (const float*)d_in[4],
          (const float*)d_in[5],  (const float*)d_in[6],  (const float*)d_in[7]};
  Head ld{(const float*)d_in[8],  (const float*)d_in[9],  (const float*)d_in[10],
          (const float*)d_in[11], (const float*)d_in[12], (const float*)d_in[13]};
  Head lo{(const float*)d_in[14], (const float*)d_in[15], (const float*)d_in[16],
          (const float*)d_in[17], (const float*)d_in[18], (const float*)d_in[19]};

  const int N = in_sizes[0] / 2;           // y is [N,2]
  hipMemsetAsync(d_out, 0, (size_t)out_size * sizeof(float), stream);

  const int rowsPerBlock = 8 * TILES_PER_WAVE * 16;   // 512
  const int blocks = N / rowsPerBlock;
  avi_kernel<<<blocks, 256, 0, stream>>>(y, zs, mu, ld, lo, (float*)d_out,
                                         1.0f / (float)N);
}